// EnhancedDLinear_71073118814416
// MI455X (gfx1250) — compile-verified
//
#include <hip/hip_runtime.h>
#include <hip/hip_bf16.h>

typedef __attribute__((ext_vector_type(2))) float v2f;
typedef __attribute__((ext_vector_type(8))) float v8f;

#define S_LEN 336
#define C_CH 96
#define B_N 8
#define N_ROWS 768   // B*C
#define SH 168
#define P_LEN 96
#define PH 48

// ---------------------------------------------------------------------------
// Stage A: xc transpose + 25-tap replicate-pad moving average decomposition
// one block per row n = b*C + c
// ---------------------------------------------------------------------------
__global__ void decomp_kernel(const float* __restrict__ x,
                              float* __restrict__ trend_in,
                              float* __restrict__ seas_in) {
  __shared__ float row[S_LEN];
  int n = blockIdx.x;
  int b = n / C_CH, c = n % C_CH;
  for (int i = threadIdx.x; i < S_LEN; i += blockDim.x)
    row[i] = x[((size_t)b * S_LEN + i) * C_CH + c];
  __syncthreads();
  for (int s = threadIdx.x; s < S_LEN; s += blockDim.x) {
    float sum = 0.f;
#pragma unroll
    for (int j = -12; j <= 12; ++j) {
      int idx = s + j;
      idx = idx < 0 ? 0 : (idx > S_LEN - 1 ? S_LEN - 1 : idx);
      sum += row[idx];
    }
    float mt = sum * (1.0f / 25.0f);
    trend_in[(size_t)n * S_LEN + s] = mt;
    seas_in[(size_t)n * S_LEN + s] = row[s] - mt;
  }
}

// ---------------------------------------------------------------------------
// WMMA f32 GEMM: out[M,Nout] = act(A[M,K] @ W[Nout,K]^T + bias)
// One wave computes a 16x16 tile with V_WMMA_F32_16X16X4_F32.
// K must be a multiple of 8; M a multiple of 16. Nout tiles padded+guarded.
// A-frag: lane(0-15)=M row, vgpr0/1 = K+0,K+1 (lanes 16-31 -> K+2,K+3)
// B-frag: lane(0-15)=N col of W row, same K striping.
// ---------------------------------------------------------------------------
__global__ __launch_bounds__(128)
void gemm_wmma_f32(const float* __restrict__ A, int lda,
                   const float* __restrict__ W, int ldw,
                   const float* __restrict__ bias,
                   float* __restrict__ Cout, int ldc,
                   int M, int Nout, int K, int numTilesN, int do_relu) {
  int wave = threadIdx.x >> 5;
  int tile = blockIdx.x * (blockDim.x >> 5) + wave;
  int numTilesM = M >> 4;
  if (tile >= numTilesM * numTilesN) return;           // wave-uniform exit
  int tm = tile / numTilesN;
  int tn = tile % numTilesN;
  int lane = threadIdx.x & 31;
  int half = lane >> 4;       // 0: K+0..1, 1: K+2..3
  int lr   = lane & 15;

  int arow = tm * 16 + lr;                              // always < M
  int bcol = tn * 16 + lr;
  int bload = bcol < Nout ? bcol : Nout - 1;            // clamp addr; cols >=Nout never stored
  const float* Ap = A + (size_t)arow * lda + 2 * half;
  const float* Wp = W + (size_t)bload * ldw + 2 * half;

  v8f acc0 = {0.f, 0.f, 0.f, 0.f, 0.f, 0.f, 0.f, 0.f};
  v8f acc1 = {0.f, 0.f, 0.f, 0.f, 0.f, 0.f, 0.f, 0.f};
  for (int k = 0; k < K; k += 8) {
    v2f a0 = *(const v2f*)(Ap + k);
    v2f b0 = *(const v2f*)(Wp + k);
    v2f a1 = *(const v2f*)(Ap + k + 4);
    v2f b1 = *(const v2f*)(Wp + k + 4);
    acc0 = __builtin_amdgcn_wmma_f32_16x16x4_f32(false, a0, false, b0,
                                                 (short)0, acc0, false, false);
    acc1 = __builtin_amdgcn_wmma_f32_16x16x4_f32(false, a1, false, b1,
                                                 (short)0, acc1, false, false);
  }
  if (bcol < Nout) {
    float bv = bias ? bias[bcol] : 0.f;
#pragma unroll
    for (int r = 0; r < 8; ++r) {
      int orow = tm * 16 + half * 8 + r;
      float v = acc0[r] + acc1[r] + bv;
      if (do_relu) v = fmaxf(v, 0.f);
      Cout[(size_t)orow * ldc + bcol] = v;
    }
  }
}

// ---------------------------------------------------------------------------
// Stage C: degenerate detail head. LayerNorm over size-1 axis => xf == ln_b,
// so detail_pred is one constant vector dp0[P]; dpm = mean(dp0).
// ---------------------------------------------------------------------------
__global__ void dp0_kernel(const float* __restrict__ ln_b,
                           const float* __restrict__ op1w,
                           const float* __restrict__ op1b,
                           const float* __restrict__ op2w,
                           const float* __restrict__ op2b,
                           float* __restrict__ dp0, float* __restrict__ dpm) {
  __shared__ float h1[SH];
  __shared__ float pv[P_LEN];
  float lb = ln_b[0];
  for (int j = threadIdx.x; j < SH; j += blockDim.x) {
    float s = 0.f;
    for (int i = 0; i < S_LEN; ++i) s += op1w[(size_t)j * S_LEN + i];
    float v = lb * s + op1b[j];
    h1[j] = v > 0.f ? v : 0.f;
  }
  __syncthreads();
  for (int p = threadIdx.x; p < P_LEN; p += blockDim.x) {
    float s = op2b[p];
    for (int j = 0; j < SH; ++j) s += h1[j] * op2w[(size_t)p * SH + j];
    dp0[p] = s;
    pv[p] = s;
  }
  __syncthreads();
  if (threadIdx.x == 0) {
    float m = 0.f;
    for (int p = 0; p < P_LEN; ++p) m += pv[p];
    dpm[0] = m / (float)P_LEN;
  }
}

// ---------------------------------------------------------------------------
// Stage D: per-(b,c) means over P of trend_pred / seasonal_pred rows
// ---------------------------------------------------------------------------
__global__ void rowmean_kernel(const float* __restrict__ tp,
                               const float* __restrict__ sp,
                               float* __restrict__ tpm, float* __restrict__ spm) {
  int n = blockIdx.x * blockDim.x + threadIdx.x;
  if (n >= N_ROWS) return;
  float s1 = 0.f, s2 = 0.f;
  for (int p = 0; p < P_LEN; ++p) {
    s1 += tp[(size_t)n * P_LEN + p];
    s2 += sp[(size_t)n * P_LEN + p];
  }
  tpm[n] = s1 / (float)P_LEN;
  spm[n] = s2 / (float)P_LEN;
}

// ---------------------------------------------------------------------------
// Stage D2: fusion weights. One block (288 thr) per batch:
// fw[b] = softmax( relu(ff @ fn1w^T + fn1b) @ fn2w^T + fn2b ) over 288
// ---------------------------------------------------------------------------
__global__ void fusionw_kernel(const float* __restrict__ tpm,
                               const float* __restrict__ spm,
                               const float* __restrict__ dpm,
                               const float* __restrict__ fn1w,
                               const float* __restrict__ fn1b,
                               const float* __restrict__ fn2w,
                               const float* __restrict__ fn2b,
                               float* __restrict__ fw) {
  const int F = 3 * C_CH, H = 32;
  __shared__ float ff[3 * C_CH];
  __shared__ float hid[32];
  __shared__ float red[3 * C_CH];
  __shared__ float smax, ssum;
  int b = blockIdx.x;
  int t = threadIdx.x;   // 0..287
  if (t < C_CH)            ff[t] = tpm[b * C_CH + t];
  else if (t < 2 * C_CH)   ff[t] = spm[b * C_CH + (t - C_CH)];
  else                     ff[t] = dpm[0];
  __syncthreads();
  if (t < H) {
    float s = fn1b[t];
    for (int i = 0; i < F; ++i) s += ff[i] * fn1w[(size_t)t * F + i];
    hid[t] = s > 0.f ? s : 0.f;
  }
  __syncthreads();
  float logit = fn2b[t];
  for (int j = 0; j < H; ++j) logit += hid[j] * fn2w[(size_t)t * H + j];
  red[t] = logit;
  __syncthreads();
  if (t == 0) {
    float m = -3.0e38f;
    for (int i = 0; i < F; ++i) m = fmaxf(m, red[i]);
    smax = m;
  }
  __syncthreads();
  float e = expf(logit - smax);
  red[t] = e;
  __syncthreads();
  if (t == 0) {
    float s = 0.f;
    for (int i = 0; i < F; ++i) s += red[i];
    ssum = s;
  }
  __syncthreads();
  fw[(size_t)b * F + t] = e / ssum;
}

// ---------------------------------------------------------------------------
// Stage E: fused[b*P+p][c] = fw0*tp + fw1*sp + fw2*dp0[p]
// ---------------------------------------------------------------------------
__global__ void fuse_kernel(const float* __restrict__ fw,
                            const float* __restrict__ tp,
                            const float* __restrict__ sp,
                            const float* __restrict__ dp0,
                            float* __restrict__ fused) {
  int id = blockIdx.x * blockDim.x + threadIdx.x;
  const int TOT = B_N * P_LEN * C_CH;
  if (id >= TOT) return;
  int c = id % C_CH;
  int m = id / C_CH;          // b*P + p
  int b = m / P_LEN, p = m % P_LEN;
  int n = b * C_CH + c;
  const float* fwb = fw + (size_t)b * (3 * C_CH);
  float v = fwb[c]              * tp[(size_t)n * P_LEN + p]
          + fwb[C_CH + c]       * sp[(size_t)n * P_LEN + p]
          + fwb[2 * C_CH + c]   * dp0[p];
  fused[(size_t)m * C_CH + c] = v;
}

// ---------------------------------------------------------------------------
extern "C" void kernel_launch(void* const* d_in, const int* in_sizes, int n_in,
                              void* d_out, int out_size, void* d_ws, size_t ws_size,
                              hipStream_t stream) {
  (void)in_sizes; (void)n_in; (void)out_size; (void)ws_size;
  const float* x    = (const float*)d_in[0];
  const float* ln_b = (const float*)d_in[22];
  const float* op1w = (const float*)d_in[23];
  const float* op1b = (const float*)d_in[24];
  const float* op2w = (const float*)d_in[25];
  const float* op2b = (const float*)d_in[26];
  const float* ls1w = (const float*)d_in[27];
  const float* ls1b = (const float*)d_in[28];
  const float* ls2w = (const float*)d_in[29];
  const float* ls2b = (const float*)d_in[30];
  const float* lt1w = (const float*)d_in[31];
  const float* lt1b = (const float*)d_in[32];
  const float* lt2w = (const float*)d_in[33];
  const float* lt2b = (const float*)d_in[34];
  const float* fn1w = (const float*)d_in[35];
  const float* fn1b = (const float*)d_in[36];
  const float* fn2w = (const float*)d_in[37];
  const float* fn2b = (const float*)d_in[38];
  const float* fp1w = (const float*)d_in[39];
  const float* fp1b = (const float*)d_in[40];
  const float* fp2w = (const float*)d_in[41];
  const float* fp2b = (const float*)d_in[42];
  float* out = (float*)d_out;

  // workspace layout (floats)
  float* ws       = (float*)d_ws;
  float* trend_in = ws;                 // 768*336
  float* seas_in  = ws + 258048;        // 768*336
  float* h_t      = ws + 516096;        // 768*168
  float* h_s      = ws + 645120;        // 768*168
  float* tp       = ws + 774144;        // 768*96
  float* sp       = ws + 847872;        // 768*96
  float* fused    = ws + 921600;        // 768*96
  float* hf       = ws + 995328;        // 768*48
  float* dp0      = ws + 1032192;       // 96 (padded 128)
  float* dpm      = ws + 1032320;       // 1  (padded 32)
  float* tpm      = ws + 1032352;       // 768
  float* spm      = ws + 1033120;       // 768
  float* fw       = ws + 1033888;       // 8*288

  // A: decomposition
  decomp_kernel<<<N_ROWS, 128, 0, stream>>>(x, trend_in, seas_in);

  // C: constant detail head (independent of stage A/B)
  dp0_kernel<<<1, 192, 0, stream>>>(ln_b, op1w, op1b, op2w, op2b, dp0, dpm);

  // B: trend / seasonal 2-layer MLPs via WMMA
  // layer1: [768,336] @ [168,336]^T -> relu -> [768,168]
  {
    int ntn = (SH + 15) / 16;                       // 11 (tail guarded)
    int tiles = (N_ROWS / 16) * ntn;                // 528
    gemm_wmma_f32<<<(tiles + 3) / 4, 128, 0, stream>>>(
        trend_in, S_LEN, lt1w, S_LEN, lt1b, h_t, SH, N_ROWS, SH, S_LEN, ntn, 1);
    gemm_wmma_f32<<<(tiles + 3) / 4, 128, 0, stream>>>(
        seas_in, S_LEN, ls1w, S_LEN, ls1b, h_s, SH, N_ROWS, SH, S_LEN, ntn, 1);
  }
  // layer2: [768,168] @ [96,168]^T -> [768,96]
  {
    int ntn = P_LEN / 16;                           // 6
    int tiles = (N_ROWS / 16) * ntn;                // 288
    gemm_wmma_f32<<<(tiles + 3) / 4, 128, 0, stream>>>(
        h_t, SH, lt2w, SH, lt2b, tp, P_LEN, N_ROWS, P_LEN, SH, ntn, 0);
    gemm_wmma_f32<<<(tiles + 3) / 4, 128, 0, stream>>>(
        h_s, SH, ls2w, SH, ls2b, sp, P_LEN, N_ROWS, P_LEN, SH, ntn, 0);
  }

  // D: channel means + fusion softmax weights
  rowmean_kernel<<<(N_ROWS + 255) / 256, 256, 0, stream>>>(tp, sp, tpm, spm);
  fusionw_kernel<<<B_N, 3 * C_CH, 0, stream>>>(tpm, spm, dpm,
                                               fn1w, fn1b, fn2w, fn2b, fw);

  // E: weighted fuse -> [768, 96]
  {
    int tot = B_N * P_LEN * C_CH;
    fuse_kernel<<<(tot + 255) / 256, 256, 0, stream>>>(fw, tp, sp, dp0, fused);
  }

  // F: final projection MLP via WMMA -> d_out [B,P,P] == [768,96] row-major
  {
    int ntn1 = PH / 16;                             // 3
    int tiles1 = (N_ROWS / 16) * ntn1;              // 144
    gemm_wmma_f32<<<(tiles1 + 3) / 4, 128, 0, stream>>>(
        fused, C_CH, fp1w, C_CH, fp1b, hf, PH, N_ROWS, PH, C_CH, ntn1, 1);
    int ntn2 = P_LEN / 16;                          // 6
    int tiles2 = (N_ROWS / 16) * ntn2;              // 288
    gemm_wmma_f32<<<(tiles2 + 3) / 4, 128, 0, stream>>>(
        hf, PH, fp2w, PH, fp2b, out, P_LEN, N_ROWS, P_LEN, PH, ntn2, 0);
  }
}